// NGCF_89902255440891
// MI455X (gfx1250) — compile-verified
//
#include <hip/hip_runtime.h>

#define D 64

typedef __attribute__((ext_vector_type(2))) float v2f;
typedef __attribute__((ext_vector_type(8))) float v8f;

// ---------------------------------------------------------------------------
// ego0 = concat(user_emb, item_emb); acc = ego0
// ---------------------------------------------------------------------------
__global__ void ngcf_init_concat(const float* __restrict__ user,
                                 const float* __restrict__ item,
                                 float* __restrict__ ego,
                                 float* __restrict__ acc,
                                 int userElems, int totalElems) {
    int i = blockIdx.x * blockDim.x + threadIdx.x;
    if (i < totalElems) {
        float v = (i < userElems) ? user[i] : item[i - userElems];
        ego[i] = v;
        acc[i] = v;
    }
}

__global__ void ngcf_zero(float* __restrict__ p, int n) {
    int i = blockIdx.x * blockDim.x + threadIdx.x;
    if (i < n) p[i] = 0.0f;
}

// ---------------------------------------------------------------------------
// COO SpMM: side[row] += val * ego[col]   (64 threads per edge, 1 dim each)
// ---------------------------------------------------------------------------
__global__ void ngcf_spmm(const int* __restrict__ rows,
                          const int* __restrict__ cols,
                          const float* __restrict__ vals,
                          const float* __restrict__ ego,
                          float* __restrict__ side,
                          int nnz) {
    long long gid = (long long)blockIdx.x * blockDim.x + threadIdx.x;
    int e = (int)(gid >> 6);
    int d = (int)(gid & 63);
    if (e >= nnz) return;
    int   r = rows[e];
    int   c = cols[e];
    float v = vals[e];
    float x = ego[(size_t)c * D + d];
    atomicAdd(&side[(size_t)r * D + d], v * x);
}

// ---------------------------------------------------------------------------
// Fused layer: H = leakyrelu(side@Wgc + bgc + (ego.*side)@Wbi + bbi, 0.2)
//              ego_out = l2norm_rows(H); acc += ego_out
// 256 threads = 8 waves; block covers 32 rows; wave w -> rowBlk=w/4 colTile=w%4
// Each wave: 16x16 f32 tile via 16 x (2 chained v_wmma_f32_16x16x4_f32)
// ---------------------------------------------------------------------------
__global__ __launch_bounds__(256)
void ngcf_layer(const float* __restrict__ ego,
                const float* __restrict__ side,
                const float* __restrict__ Wgc,
                const float* __restrict__ Wbi,
                const float* __restrict__ bgc,
                const float* __restrict__ bbi,
                float* __restrict__ ego_out,
                float* __restrict__ acc,
                int nrows) {
    __shared__ float sWgc[D * D];     // 16 KB
    __shared__ float sWbi[D * D];     // 16 KB
    __shared__ float sBias[D];
    __shared__ float sTile[32 * D];   // 8 KB post-activation tile
    __shared__ float sRnorm[32];

    const int tid = threadIdx.x;

    // Stage weights + combined bias in LDS
    for (int i = tid; i < D * D; i += 256) {
        sWgc[i] = Wgc[i];
        sWbi[i] = Wbi[i];
    }
    if (tid < D) sBias[tid] = bgc[tid] + bbi[tid];
    __syncthreads();

    const int lane    = tid & 31;
    const int wave    = tid >> 5;     // 0..7
    const int rowBlk  = wave >> 2;    // 0..1
    const int colTile = wave & 3;     // 0..3

    const int m  = lane & 15;         // M within tile
    const int hi = lane >> 4;         // lane group -> K pair select
    int row = blockIdx.x * 32 + rowBlk * 16 + m;
    if (row >= nrows) row = nrows - 1;            // defensive clamp, no EXEC change
    const int n = colTile * 16 + (lane & 15);     // N within 64 cols

    const float* sideRow = side + (size_t)row * D;
    const float* egoRow  = ego  + (size_t)row * D;

    v8f cacc = {0.f, 0.f, 0.f, 0.f, 0.f, 0.f, 0.f, 0.f};

#pragma unroll
    for (int kk = 0; kk < D; kk += 4) {
        const int kb = kk + hi * 2;   // lanes 0-15: K={kk,kk+1}; lanes 16-31: K={kk+2,kk+3}
        float2 s2 = *(const float2*)(sideRow + kb);
        float2 e2 = *(const float2*)(egoRow + kb);
        v2f aGc; aGc.x = s2.x;        aGc.y = s2.y;
        v2f aBi; aBi.x = s2.x * e2.x; aBi.y = s2.y * e2.y;
        v2f bGc; bGc.x = sWgc[(kb + 0) * D + n]; bGc.y = sWgc[(kb + 1) * D + n];
        v2f bBi; bBi.x = sWbi[(kb + 0) * D + n]; bBi.y = sWbi[(kb + 1) * D + n];
        cacc = __builtin_amdgcn_wmma_f32_16x16x4_f32(false, aGc, false, bGc,
                                                     (short)0, cacc, false, false);
        cacc = __builtin_amdgcn_wmma_f32_16x16x4_f32(false, aBi, false, bBi,
                                                     (short)0, cacc, false, false);
    }

    // Epilogue: bias + leaky-relu -> LDS tile (C/D layout: VGPR v -> M = v + 8*hi)
    const float bias = sBias[n];
    const int rowLocalBase = rowBlk * 16 + hi * 8;
#pragma unroll
    for (int v = 0; v < 8; ++v) {
        float val = cacc[v] + bias;
        val = (val > 0.f) ? val : 0.2f * val;
        sTile[(rowLocalBase + v) * D + n] = val;
    }
    __syncthreads();

    // Per-row L2 norm (32 rows, one thread each)
    if (tid < 32) {
        float ss = 0.f;
        const float4* rp = (const float4*)(sTile + tid * D);
#pragma unroll
        for (int i = 0; i < D / 4; ++i) {
            float4 q = rp[i];
            ss += q.x * q.x + q.y * q.y + q.z * q.z + q.w * q.w;
        }
        float nm = sqrtf(ss);
        sRnorm[tid] = 1.f / fmaxf(nm, 1e-12f);
    }
    __syncthreads();

    // Normalize, write ego_out, accumulate into acc
    const int gBase = blockIdx.x * 32 * D;
#pragma unroll
    for (int i = 0; i < 8; ++i) {
        int idx = tid + i * 256;          // 0..2047
        int r   = idx >> 6;               // local row
        if (blockIdx.x * 32 + r < nrows) {
            float val = sTile[idx] * sRnorm[r];
            int g = gBase + idx;
            ego_out[g] = val;
            acc[g] += val;
        }
    }
}

__global__ void ngcf_final(const float* __restrict__ acc,
                           float* __restrict__ out,
                           float scale, int n) {
    int i = blockIdx.x * blockDim.x + threadIdx.x;
    if (i < n) out[i] = acc[i] * scale;
}

// ---------------------------------------------------------------------------
// Host-side orchestration (stream-capture safe: only kernel launches)
// ---------------------------------------------------------------------------
extern "C" void kernel_launch(void* const* d_in, const int* in_sizes, int n_in,
                              void* d_out, int out_size, void* d_ws, size_t ws_size,
                              hipStream_t stream) {
    const float* user = (const float*)d_in[0];
    const float* item = (const float*)d_in[1];
    const float* Wgc  = (const float*)d_in[2];
    const float* Wbi  = (const float*)d_in[3];
    const float* bgc  = (const float*)d_in[4];
    const float* bbi  = (const float*)d_in[5];
    const int*   rows = (const int*)d_in[6];
    const int*   cols = (const int*)d_in[7];
    const float* vals = (const float*)d_in[8];

    const int userElems = in_sizes[0];
    const int itemElems = in_sizes[1];
    const int N   = (userElems + itemElems) / D;   // 100000
    const int L   = in_sizes[2] / (D * D);         // 3
    const int nnz = in_sizes[6];                   // 3200000
    const int ND  = N * D;

    float* ws   = (float*)d_ws;
    size_t nd   = (size_t)ND;
    float* egoA = ws;
    float* egoB = ws + nd;
    float* side = ws + 2 * nd;
    float* acc  = ws + 3 * nd;

    const int elemBlocks = (ND + 255) / 256;

    ngcf_init_concat<<<elemBlocks, 256, 0, stream>>>(user, item, egoA, acc, userElems, ND);

    float* cur = egoA;
    float* nxt = egoB;
    for (int k = 0; k < L; ++k) {
        ngcf_zero<<<elemBlocks, 256, 0, stream>>>(side, ND);

        long long spmmThreads = (long long)nnz * 64;
        int spmmBlocks = (int)((spmmThreads + 255) / 256);
        ngcf_spmm<<<spmmBlocks, 256, 0, stream>>>(rows, cols, vals, cur, side, nnz);

        int layerBlocks = (N + 31) / 32;
        ngcf_layer<<<layerBlocks, 256, 0, stream>>>(cur, side,
                                                    Wgc + (size_t)k * D * D,
                                                    Wbi + (size_t)k * D * D,
                                                    bgc + (size_t)k * D,
                                                    bbi + (size_t)k * D,
                                                    nxt, acc, N);
        float* t = cur; cur = nxt; nxt = t;
    }

    float scale = 1.0f / (float)(L + 1);
    ngcf_final<<<elemBlocks, 256, 0, stream>>>(acc, (float*)d_out, scale, ND);
}